// CTCLoss_83777632076148
// MI455X (gfx1250) — compile-verified
//
#include <hip/hip_runtime.h>
#include <cstdint>
#include <cstddef>

// ---------------------------------------------------------------------------
// CTC loss forward, CDNA5 (gfx1250).
// One workgroup per batch element; 544 threads (17 wave32s) cover S=513 states.
// alpha ping-pong in LDS. Per-timestep probability row (512 B) is prefetched
// 2-deep into LDS with global_load_async_to_lds_b128 (wave 0, 32 x b128) and
// synced with s_wait_asynccnt; a pipelined stage (threads 0..127) converts the
// resident row t+1 to log-space while all 513 state-threads run DP step t, so
// the DP critical path has no logf — just one LDS read per lp.
// ---------------------------------------------------------------------------

constexpr int B_ = 64;
constexpr int T_ = 1024;
constexpr int C_ = 128;
constexpr int L_ = 256;
constexpr int S_ = 2 * L_ + 1;   // 513
constexpr int BLANK_ = C_ - 1;   // 127
constexpr int NTH = 544;         // 17 waves of 32

#define NEGV (-1e30f)
#define EPSV (1e-7f)

#if defined(__HIP_DEVICE_COMPILE__) && __has_builtin(__builtin_amdgcn_global_load_async_to_lds_b128)
#define HAVE_ASYNC 1
#else
#define HAVE_ASYNC 0
#endif

#if HAVE_ASYNC
typedef int v4i_t __attribute__((__vector_size__(4 * sizeof(int))));
typedef __attribute__((address_space(1))) v4i_t* g_v4i_p;
typedef __attribute__((address_space(3))) v4i_t* l_v4i_p;
#endif

__device__ __forceinline__ void async_row_load16(const float* gsrc, float* ldst) {
#if HAVE_ASYNC
  // Flat LDS addresses carry the LDS byte offset in the low 32 bits (ISA §10.2
  // aperture mapping), so integer casts to AS(1)/AS(3) pointers are well-defined
  // and sidestep cross-AS pointer-cast / constness restrictions.
  g_v4i_p g1 = (g_v4i_p)(uintptr_t)(const void*)gsrc;
  l_v4i_p l3 = (l_v4i_p)(uint32_t)(uintptr_t)(void*)ldst;
  __builtin_amdgcn_global_load_async_to_lds_b128(g1, l3, 0, 0);
#else
  const float4* s4 = reinterpret_cast<const float4*>(gsrc);
  float4* d4 = reinterpret_cast<float4*>(ldst);
  *d4 = *s4;
#endif
}

template <int N>
__device__ __forceinline__ void wait_async_le() {
#if HAVE_ASYNC
#if __has_builtin(__builtin_amdgcn_s_wait_asynccnt)
  __builtin_amdgcn_s_wait_asynccnt(N);
#else
  asm volatile("s_wait_asynccnt %0" ::"i"(N) : "memory");
#endif
#endif
}

__global__ __launch_bounds__(NTH) void ctc_loss_kernel(
    const int* __restrict__ y_true,        // (B, L)
    const float* __restrict__ y_pred,      // (B, T, C)
    const int* __restrict__ input_length,  // (B, 1)
    const int* __restrict__ label_length,  // (B, 1)
    float* __restrict__ out)               // (B, 1)
{
  __shared__ __align__(16) float alphaA[S_ + 2];   // [0..1] = NEG pad
  __shared__ __align__(16) float alphaB[S_ + 2];
  __shared__ __align__(16) float rawp[2][C_];      // raw prob rows (row r -> rawp[r&1])
  __shared__ __align__(16) float lgp[2][C_];       // log-prob rows  (row r -> lgp[r&1])

  const int tid = threadIdx.x;
  const int b = blockIdx.x;
  const int lab_len = label_length[b];
  const int in_len = input_length[b];

  const float* rowbase = y_pred + (size_t)b * T_ * C_;

  // ---- prologue: rows 0 and 1 resident --------------------------------------
  if (tid < 32) {
    async_row_load16(rowbase + 0 * C_ + tid * 4, &rawp[0][tid * 4]);
    async_row_load16(rowbase + 1 * C_ + tid * 4, &rawp[1][tid * 4]);
    wait_async_le<0>();
  }
  if (tid < 2) { alphaA[tid] = NEGV; alphaB[tid] = NEGV; }
  __syncthreads();  // rows 0,1 visible to all

  // ---- per-thread state setup ------------------------------------------------
  const int s = tid;
  int ext_s = BLANK_;
  bool skip_ok = false;
  bool valid = false;
  float a_cur = NEGV;

  if (s < S_) {
    if (s & 1) {
      ext_s = y_true[b * L_ + (s >> 1)];
      if (s >= 3) {
        const int prev = y_true[b * L_ + ((s - 3) >> 1)];
        skip_ok = (ext_s != BLANK_) && (ext_s != prev);
      }
    }
    valid = (s < 2 * lab_len + 1);
    if (s == 0) {
      a_cur = logf(rawp[0][BLANK_] + EPSV);
    } else if (s == 1 && lab_len > 0) {
      a_cur = logf(rawp[0][ext_s] + EPSV);
    }
    if (!valid) a_cur = NEGV;
  }

  // Convert row 1 to log-space (needed by DP step t=1).
  if (tid < C_) lgp[1][tid] = logf(rawp[1][tid] + EPSV);

  float* cur = alphaA;
  float* nxt = alphaB;
  if (s < S_) cur[2 + s] = a_cur;

  __syncthreads();  // conversions/alpha0 done; rawp[0] free for reuse

  // Row 2 -> rawp[0] (prefetch depth 2 from here on).
  if (tid < 32) {
    async_row_load16(rowbase + 2 * C_ + tid * 4, &rawp[0][tid * 4]);
  }

  // ---- main DP loop over time ------------------------------------------------
  for (int t = 1; t < T_; ++t) {
    if (tid < 32) {
      if (t + 2 < T_) {
        // Row t+2 -> rawp[t&1] (its old content, row t, was consumed at t-1).
        async_row_load16(rowbase + (size_t)(t + 2) * C_ + tid * 4,
                         &rawp[t & 1][tid * 4]);
        wait_async_le<1>();   // in-order completion => row t+1 resident
      } else {
        wait_async_le<0>();   // drain: all issued rows resident
      }
    }
    __syncthreads();          // (A) row t+1 raw + lgp[t&1] + prev alpha visible

    // Pipelined log conversion of row t+1 (4 waves), off the DP critical path.
    if (t + 1 < T_ && tid < C_) {
      lgp[(t + 1) & 1][tid] = logf(rawp[(t + 1) & 1][tid] + EPSV);
    }

    // DP step t for state s.
    float a_new = a_cur;
    if (s < S_ && valid) {
      const float a0 = a_cur;
      const float a1 = cur[2 + s - 1];
      const float a2 = skip_ok ? cur[2 + s - 2] : NEGV;
      const float m = fmaxf(a0, fmaxf(a1, a2));
      const float sum = __expf(a0 - m) + __expf(a1 - m) + __expf(a2 - m);
      const float v = m + __logf(sum) + lgp[t & 1][ext_s];
      a_new = (t < in_len) ? v : a_cur;
    }
    if (s < S_) nxt[2 + s] = a_new;
    a_cur = a_new;
    { float* tmp = cur; cur = nxt; nxt = tmp; }

    __syncthreads();          // (B) raw/log buffer reuse + alpha ping-pong safe
  }

  // ---- epilogue: loss = -logaddexp(alpha[2l], alpha[2l-1]) -------------------
  if (tid == 0) {
    const float eb = cur[2 + 2 * lab_len];
    const float el = (lab_len > 0) ? cur[2 + 2 * lab_len - 1] : NEGV;
    const float m = fmaxf(eb, el);
    const float loss = -(m + logf(__expf(eb - m) + __expf(el - m)));
    out[b] = loss;
  }
}

extern "C" void kernel_launch(void* const* d_in, const int* in_sizes, int n_in,
                              void* d_out, int out_size, void* d_ws, size_t ws_size,
                              hipStream_t stream) {
  const int*   y_true       = (const int*)d_in[0];
  const float* y_pred       = (const float*)d_in[1];
  const int*   input_length = (const int*)d_in[2];
  const int*   label_length = (const int*)d_in[3];
  float*       out          = (float*)d_out;

  ctc_loss_kernel<<<B_, NTH, 0, stream>>>(y_true, y_pred, input_length,
                                          label_length, out);
}